// BitflipByProbabilityLayer_12154757448296
// MI455X (gfx1250) — compile-verified
//
#include <hip/hip_runtime.h>
#include <stdint.h>

// ---------------------------------------------------------------------------
// BitflipByProbabilityLayer for MI455X (gfx1250, wave32).
//
// Memory-bound: 64 MiB in + 64 MiB out = 134 MB -> ~5.8 us at 23.3 TB/s.
// No matrix ops in the reference -> no WMMA. Optimization levers:
//   * b128 global load/store with non-temporal hints (th:TH_LOAD_NT /
//     th:TH_STORE_NT, gfx1250 cache-policy) -- data touched exactly once.
//   * Keep VALU far below the memory-bound budget: instead of 32 Bernoulli
//     draws per word, draw the flip COUNT once per 128-bit vector,
//     k ~ Binomial(128, 0.001), by inverse CDF from one counter-hash:
//       P(k=0)  = 0.999^128 = 0.8797970
//       P(k<=1) =             0.9925238
//       P(k<=2) =             0.9996891
//       P(k<=3) =             0.9999903   (k>=5 residual 2.6e-7, truncated)
//     then place k bits at hashed 7-bit positions (64-bit halves, XOR).
//   ~9 VALU ops per 16 bytes on the fast path; placements are EXEC-masked
//   and the 2nd/3rd run in only ~21% / ~1% of wave-iterations.
// Deterministic: stateless counter hash keyed on the vector index.
// ---------------------------------------------------------------------------

typedef uint32_t u32;
typedef unsigned long long u64;
typedef u32 u32x4 __attribute__((ext_vector_type(4)));

__device__ __forceinline__ u32 lowbias32(u32 x) {
    x ^= x >> 16; x *= 0x7feb352dU;
    x ^= x >> 15; x *= 0x846ca68bU;
    x ^= x >> 16;
    return x;
}

// XOR one bit (position 0..127) into a 128-bit value held as two 64-bit halves.
__device__ __forceinline__ void place_bit(u64& lo, u64& hi, u32 pos) {
    u64 b = 1ull << (pos & 63u);
    if (pos & 64u) hi ^= b; else lo ^= b;   // -> cndmask pair, branchless
}

__device__ __forceinline__ constexpr u32 SEED() { return 0x5851F42Du ^ 42u; }

// Inverse-CDF thresholds for Binomial(128, 0.001), scaled to 2^32.
__device__ __forceinline__ constexpr u32 TB0() { return (u32)(0.8797970 * 4294967296.0); }
__device__ __forceinline__ constexpr u32 TB1() { return (u32)(0.9925238 * 4294967296.0); }
__device__ __forceinline__ constexpr u32 TB2() { return (u32)(0.9996891 * 4294967296.0); }
__device__ __forceinline__ constexpr u32 TB3() { return (u32)(0.9999903 * 4294967296.0); }

__global__ __launch_bounds__(256)
void bitflip_b128_kernel(const u32x4* __restrict__ in, u32x4* __restrict__ out, u32 n4) {
    u32 tid    = blockIdx.x * blockDim.x + threadIdx.x;
    u32 stride = gridDim.x * blockDim.x;
    for (u32 i = tid; i < n4; i += stride) {
        // Streaming b128, read-once / write-once -> non-temporal policy.
        u32x4 v = __builtin_nontemporal_load(in + i);

        // One uniform draw per 128 bits.
        u32 u = lowbias32(i * 0x9E3779B1u + SEED());

        if (__builtin_expect(u >= TB0(), 0)) {      // ~12% of vectors: k >= 1
            // Fresh scramble for positions (4 x 7 bits available).
            u32 h = lowbias32(u ^ 0x68E31DA4u);
            u64 mlo = 0, mhi = 0;
            place_bit(mlo, mhi, h & 127u);
            if (u >= TB1()) {                       // k >= 2 (~0.75% of vectors)
                place_bit(mlo, mhi, (h >> 7) & 127u);
                if (u >= TB2()) {                   // k >= 3
                    place_bit(mlo, mhi, (h >> 14) & 127u);
                    if (u >= TB3())                 // k >= 4
                        place_bit(mlo, mhi, (h >> 21) & 127u);
                }
            }
            v.x ^= (u32)mlo;  v.y ^= (u32)(mlo >> 32);
            v.z ^= (u32)mhi;  v.w ^= (u32)(mhi >> 32);
        }

        __builtin_nontemporal_store(v, out + i);
    }
}

// Scalar tail (n not divisible by 4): per-element Binomial(32, 0.001).
__device__ __forceinline__ u32 flip_mask32(u32 idx) {
    constexpr u32 T0 = (u32)(0.96849100 * 4294967296.0);
    constexpr u32 T1 = (u32)(0.99951374 * 4294967296.0);
    constexpr u32 T2 = (u32)(0.99999507 * 4294967296.0);
    u32 u = lowbias32(idx * 0x9E3779B1u + SEED());
    u32 mask = 0u;
    if (u >= T0) {
        u32 h = lowbias32(u ^ 0x85EBCA77u);
        mask = 1u << (h & 31u);
        if (u >= T1) mask ^= 1u << ((h >> 5)  & 31u);
        if (u >= T2) mask ^= 1u << ((h >> 10) & 31u);
    }
    return mask;
}

__global__ void bitflip_tail_kernel(const u32* __restrict__ in, u32* __restrict__ out,
                                    u32 start, u32 n) {
    u32 i = start + blockIdx.x * blockDim.x + threadIdx.x;
    if (i < n) {
        u32 v = __builtin_nontemporal_load(in + i);
        __builtin_nontemporal_store(v ^ flip_mask32(i), out + i);
    }
}

extern "C" void kernel_launch(void* const* d_in, const int* in_sizes, int n_in,
                              void* d_out, int out_size, void* d_ws, size_t ws_size,
                              hipStream_t stream) {
    (void)in_sizes; (void)n_in; (void)d_ws; (void)ws_size;
    const u32* x = (const u32*)d_in[0];   // float32 handled bitwise
    u32*       y = (u32*)d_out;

    u32 n  = (u32)out_size;               // 16*1024*1024 for the reference shape
    u32 n4 = n >> 2;                      // number of 128-bit vectors

    if (n4) {
        const int threads = 256;          // 8 wave32s per block
        u32 blocks = (n4 + (u32)threads - 1u) / (u32)threads;
        if (blocks > (1u << 20)) blocks = (1u << 20);  // grid-stride covers rest
        bitflip_b128_kernel<<<dim3(blocks), dim3(threads), 0, stream>>>(
            (const u32x4*)x, (u32x4*)y, n4);
    }
    u32 rem = n & 3u;
    if (rem) {
        bitflip_tail_kernel<<<dim3(1), dim3(32), 0, stream>>>(x, y, n4 << 2, n);
    }
}